// miniBatchOT_74758200754730
// MI455X (gfx1250) — compile-verified
//
#include <hip/hip_runtime.h>
#include <hip/hip_bf16.h>

// ---------------------------------------------------------------------------
// Mini-batch OT loss (Sinkhorn) for MI455X / gfx1250, wave32.
//   1) gather_kernel   : build F,Z [4096,1024], row sq-norms, init log_v, out=0
//   2) gemm_cost       : cost = 0.5*max(fn+zn-2*F.Z^T, 0)
//                        v_wmma_f32_16x16x4_f32 + double-buffered
//                        global_load_async_to_lds_b128 staging (ASYNCcnt)
//   3) 50x { row_lse ; col_part ; col_comb } : log-domain Sinkhorn, b128
//                        nontemporal streaming from L2-resident cost matrix
//   4) loss_kernel     : sum( exp(lu+mK+lv) * cost ) -> atomic add into d_out
// ---------------------------------------------------------------------------

typedef float v2f __attribute__((ext_vector_type(2)));
typedef float v4f __attribute__((ext_vector_type(4)));
typedef float v8f __attribute__((ext_vector_type(8)));

#define BATCH   4096
#define DHALF   512
#define DCAT    1024
#define NMAT    4096
#define INV_EPS 20.0f            // 1/0.05
#define LOG_AB  (-8.317766167f)  // -log(4096)
#define NEG_INF (-__builtin_huge_valf())

// One-exp online logsumexp update.
__device__ __forceinline__ void lse_upd(float x, float& m, float& s) {
    if (x > m) { s = s * __expf(m - x) + 1.0f; m = x; }
    else       { s += __expf(x - m); }
}
// Combine two (max, sumexp) partials.
__device__ __forceinline__ void lse_comb(float m2, float s2, float& m, float& s) {
    const float M = fmaxf(m, m2);
    s = s * __expf(m - M) + s2 * __expf(m2 - M);
    m = M;
}
// gfx1250 async DMA: 16B global -> LDS, tracked by ASYNCcnt.
__device__ __forceinline__ void async_copy16(unsigned lds_off, const float* gptr) {
    asm volatile("global_load_async_to_lds_b128 %0, %1, off"
                 :: "v"(lds_off), "v"((unsigned long long)(size_t)gptr)
                 : "memory");
}

// ---------------- Phase 1: gather + concat + row norms ----------------------
__global__ __launch_bounds__(256) void ot_gather_kernel(
    const float* __restrict__ feature, const float* __restrict__ zero_feature,
    const float* __restrict__ text, const float* __restrict__ zero_text,
    const int* __restrict__ gt,
    float* __restrict__ F, float* __restrict__ Z,
    float* __restrict__ fn, float* __restrict__ zn,
    float* __restrict__ log_v, float* __restrict__ out)
{
    __shared__ float sf[256], sz[256];
    const int i   = blockIdx.x;
    const int tid = threadIdx.x;
    const int g   = gt[i];

    float accF = 0.f, accZ = 0.f;
    for (int k = tid; k < DHALF; k += 256) {
        float a  = feature[i * DHALF + k];
        float t  = text[g * DHALF + k];
        float zb = zero_feature[i * DHALF + k];   // [B,1,D] flat == [B,D]
        float zt = zero_text[g * DHALF + k];
        F[i * DCAT + k]         = a;
        F[i * DCAT + DHALF + k] = t;
        Z[i * DCAT + k]         = zb;
        Z[i * DCAT + DHALF + k] = zt;
        accF += a * a + t * t;
        accZ += zb * zb + zt * zt;
    }
    sf[tid] = accF; sz[tid] = accZ;
    __syncthreads();
    for (int w = 128; w > 0; w >>= 1) {
        if (tid < w) { sf[tid] += sf[tid + w]; sz[tid] += sz[tid + w]; }
        __syncthreads();
    }
    if (tid == 0) {
        fn[i] = sf[0];
        zn[i] = sz[0];
        log_v[i] = 0.0f;
        if (i == 0) out[0] = 0.0f;
    }
}

// ---------------- Phase 2: cost GEMM, f32 WMMA + async LDS staging ----------
// Block tile 64(M) x 32(N), K-tile 32, 8 waves (4x2), 16x16 per wave.
// LDS rows padded to 36 floats: 16B-aligned rows for b128 async writes and
// 16 distinct banks across the 16-lane fragment groups.
__global__ __launch_bounds__(256) void ot_gemm_cost_kernel(
    const float* __restrict__ F, const float* __restrict__ Z,
    const float* __restrict__ fn, const float* __restrict__ zn,
    float* __restrict__ Cost)
{
    __shared__ float As[2][64][36];
    __shared__ float Bs[2][32][36];

    const int tid  = threadIdx.x;
    const int m0   = blockIdx.y * 64;
    const int n0   = blockIdx.x * 32;
    const int lane = tid & 31;
    const int wave = tid >> 5;
    const int wm   = wave & 3;           // M-wave 0..3
    const int wn   = wave >> 2;          // N-wave 0..1
    const int half = lane >> 4;          // A/B: 0 -> K0/K1, 1 -> K2/K3
    const int lm   = lane & 15;          // M (A frag) / N (B frag)

    const int lr = tid >> 3;             // staging row 0..31
    const int lc = (tid & 7) * 4;        // staging col group (float4)

    const float* gA0 = F + (size_t)(m0 + lr)      * DCAT + lc;
    const float* gA1 = F + (size_t)(m0 + lr + 32) * DCAT + lc;
    const float* gB  = Z + (size_t)(n0 + lr)      * DCAT + lc;

    // Prologue: async-fill buffer 0 for K-tile 0.
    async_copy16((unsigned)(size_t)&As[0][lr][lc],      gA0);
    async_copy16((unsigned)(size_t)&As[0][lr + 32][lc], gA1);
    async_copy16((unsigned)(size_t)&Bs[0][lr][lc],      gB);

    v8f acc = {};
    for (int k0 = 0; k0 < DCAT; k0 += 32) {
        const int buf = (k0 >> 5) & 1;
        if (k0 + 32 < DCAT) {
            const int nb = buf ^ 1;      // prefetch next tile while computing
            async_copy16((unsigned)(size_t)&As[nb][lr][lc],      gA0 + k0 + 32);
            async_copy16((unsigned)(size_t)&As[nb][lr + 32][lc], gA1 + k0 + 32);
            async_copy16((unsigned)(size_t)&Bs[nb][lr][lc],      gB  + k0 + 32);
            asm volatile("s_wait_asynccnt 0x3" ::: "memory");  // tile k0 done
        } else {
            asm volatile("s_wait_asynccnt 0x0" ::: "memory");
        }
        __syncthreads();

        #pragma unroll
        for (int kk = 0; kk < 32; kk += 4) {
            const int ka = kk + 2 * half;
            v2f a, b;
            a.x = As[buf][wm * 16 + lm][ka];
            a.y = As[buf][wm * 16 + lm][ka + 1];
            b.x = Bs[buf][wn * 16 + lm][ka];     // B[k][n] = Z[n][k]
            b.y = Bs[buf][wn * 16 + lm][ka + 1];
            acc = __builtin_amdgcn_wmma_f32_16x16x4_f32(
                false, a, false, b, (short)0, acc, false, false);
        }
        __syncthreads();
    }

    // Epilogue: cost = 0.5*max(fn+zn-2*dot, 0).
    const int col = n0 + wn * 16 + lm;
    const float znv = zn[col];
    #pragma unroll
    for (int v = 0; v < 8; ++v) {
        const int row = m0 + wm * 16 + v + 8 * half;
        const float c = 0.5f * fmaxf(fn[row] + znv - 2.0f * acc[v], 0.0f);
        Cost[(size_t)row * NMAT + col] = c;
    }
}

// ---------------- Phase 3a: row logsumexp (update log_u) --------------------
// One block per row, b128 nontemporal streaming, per-thread online LSE.
__global__ __launch_bounds__(256) void ot_row_lse_kernel(
    const float* __restrict__ Cost, const float* __restrict__ log_v,
    float* __restrict__ log_u)
{
    __shared__ float sm[256], ss[256];
    const int i   = blockIdx.x;
    const int tid = threadIdx.x;
    const v4f* row4 = (const v4f*)(Cost + (size_t)i * NMAT);
    const v4f* lv4  = (const v4f*)log_v;

    float m = NEG_INF, s = 0.0f;
    #pragma unroll
    for (int j4 = tid; j4 < NMAT / 4; j4 += 256) {
        const v4f c = __builtin_nontemporal_load(row4 + j4);
        const v4f v = lv4[j4];
        lse_upd(v.x - c.x * INV_EPS, m, s);
        lse_upd(v.y - c.y * INV_EPS, m, s);
        lse_upd(v.z - c.z * INV_EPS, m, s);
        lse_upd(v.w - c.w * INV_EPS, m, s);
    }
    sm[tid] = m; ss[tid] = s;
    __syncthreads();
    for (int w = 128; w > 0; w >>= 1) {
        if (tid < w) {
            float m1 = sm[tid], s1 = ss[tid];
            lse_comb(sm[tid + w], ss[tid + w], m1, s1);
            sm[tid] = m1; ss[tid] = s1;
        }
        __syncthreads();
    }
    if (tid == 0) log_u[i] = LOG_AB - (sm[0] + __logf(ss[0]));
}

// ---------------- Phase 3b: column logsumexp, stage 1 (partials) ------------
// Grid (32 col-blocks, 8 row-chunks). Block = 32 col-groups x 8 row-strides;
// each thread owns 4 consecutive columns -> 512B contiguous per wave per row.
__global__ __launch_bounds__(256) void ot_col_part_kernel(
    const float* __restrict__ Cost, const float* __restrict__ log_u,
    float* __restrict__ pm, float* __restrict__ ps)
{
    __shared__ float sm[8][32][4], ss[8][32][4];
    const int tx   = threadIdx.x & 31;
    const int ty   = threadIdx.x >> 5;
    const int col0 = (blockIdx.x * 32 + tx) * 4;
    const int r0   = blockIdx.y * 512;

    float m[4] = {NEG_INF, NEG_INF, NEG_INF, NEG_INF};
    float s[4] = {0.f, 0.f, 0.f, 0.f};

    for (int r = r0 + ty; r < r0 + 512; r += 8) {
        const v4f c = __builtin_nontemporal_load(
            (const v4f*)(Cost + (size_t)r * NMAT + col0));
        const float lu = log_u[r];
        lse_upd(lu - c.x * INV_EPS, m[0], s[0]);
        lse_upd(lu - c.y * INV_EPS, m[1], s[1]);
        lse_upd(lu - c.z * INV_EPS, m[2], s[2]);
        lse_upd(lu - c.w * INV_EPS, m[3], s[3]);
    }
    #pragma unroll
    for (int c = 0; c < 4; ++c) { sm[ty][tx][c] = m[c]; ss[ty][tx][c] = s[c]; }
    __syncthreads();
    if (ty == 0) {
        #pragma unroll
        for (int c = 0; c < 4; ++c) {
            float M = sm[0][tx][c], S = ss[0][tx][c];
            #pragma unroll
            for (int t = 1; t < 8; ++t) lse_comb(sm[t][tx][c], ss[t][tx][c], M, S);
            pm[(size_t)blockIdx.y * NMAT + col0 + c] = M;
            ps[(size_t)blockIdx.y * NMAT + col0 + c] = S;
        }
    }
}

// ---------------- Phase 3b: column logsumexp, stage 2 (combine) -------------
__global__ __launch_bounds__(256) void ot_col_comb_kernel(
    const float* __restrict__ pm, const float* __restrict__ ps,
    float* __restrict__ log_v)
{
    const int j = blockIdx.x * 256 + threadIdx.x;     // 0..4095
    float m = pm[j], s = ps[j];
    #pragma unroll
    for (int c = 1; c < 8; ++c) lse_comb(pm[(size_t)c * NMAT + j],
                                         ps[(size_t)c * NMAT + j], m, s);
    log_v[j] = LOG_AB - (m + __logf(s));
}

// ---------------- Phase 4: loss = sum(pi * cost) ----------------------------
__global__ __launch_bounds__(256) void ot_loss_kernel(
    const float* __restrict__ Cost, const float* __restrict__ log_u,
    const float* __restrict__ log_v, float* __restrict__ out)
{
    __shared__ float sd[256];
    const int tid = threadIdx.x;
    const size_t total4  = (size_t)NMAT * NMAT / 4;
    const size_t stride  = (size_t)gridDim.x * 256;
    const v4f* Cost4 = (const v4f*)Cost;
    const v4f* lv4   = (const v4f*)log_v;

    float acc = 0.0f;
    for (size_t idx4 = (size_t)blockIdx.x * 256 + tid; idx4 < total4; idx4 += stride) {
        const int i  = (int)(idx4 >> 10);
        const int j4 = (int)(idx4 & (NMAT / 4 - 1));
        const v4f c = __builtin_nontemporal_load(Cost4 + idx4);
        const v4f v = lv4[j4];
        const float lu = log_u[i];
        acc += c.x * __expf(lu + v.x - c.x * INV_EPS);
        acc += c.y * __expf(lu + v.y - c.y * INV_EPS);
        acc += c.z * __expf(lu + v.z - c.z * INV_EPS);
        acc += c.w * __expf(lu + v.w - c.w * INV_EPS);
    }
    sd[tid] = acc;
    __syncthreads();
    for (int w = 128; w > 0; w >>= 1) {
        if (tid < w) sd[tid] += sd[tid + w];
        __syncthreads();
    }
    if (tid == 0) atomicAdd(out, sd[0]);
}

// ---------------------------------------------------------------------------
extern "C" void kernel_launch(void* const* d_in, const int* in_sizes, int n_in,
                              void* d_out, int out_size, void* d_ws, size_t ws_size,
                              hipStream_t stream)
{
    const float* feature      = (const float*)d_in[0];
    const float* zero_feature = (const float*)d_in[1];
    const float* text         = (const float*)d_in[2];
    const float* zero_text    = (const float*)d_in[3];
    const int*   groundtruth  = (const int*)d_in[4];
    float* out = (float*)d_out;

    // Workspace layout (~96.4 MiB); every block offset is a multiple of
    // 4096 floats -> 16B alignment for v4f loads everywhere.
    float* F    = (float*)d_ws;
    float* Z    = F + (size_t)BATCH * DCAT;
    float* Cost = Z + (size_t)BATCH * DCAT;
    float* fn   = Cost + (size_t)NMAT * NMAT;
    float* zn   = fn + BATCH;
    float* lu   = zn + BATCH;
    float* lv   = lu + BATCH;
    float* pm   = lv + BATCH;                 // [8][4096] col-pass partial max
    float* ps   = pm + 8 * (size_t)NMAT;      // [8][4096] col-pass partial sum

    ot_gather_kernel<<<BATCH, 256, 0, stream>>>(
        feature, zero_feature, text, zero_text, groundtruth,
        F, Z, fn, zn, lv, out);

    dim3 ggrid(NMAT / 32, BATCH / 64);
    ot_gemm_cost_kernel<<<ggrid, 256, 0, stream>>>(F, Z, fn, zn, Cost);

    for (int it = 0; it < 50; ++it) {
        ot_row_lse_kernel<<<NMAT, 256, 0, stream>>>(Cost, lv, lu);
        ot_col_part_kernel<<<dim3(32, 8), 256, 0, stream>>>(Cost, lu, pm, ps);
        ot_col_comb_kernel<<<NMAT / 256, 256, 0, stream>>>(pm, ps, lv);
    }

    ot_loss_kernel<<<1024, 256, 0, stream>>>(Cost, lu, lv, out);
}